// GBPConvLayer_68616397521436
// MI455X (gfx1250) — compile-verified
//
#include <hip/hip_runtime.h>

typedef __attribute__((ext_vector_type(2))) float v2f;
typedef __attribute__((ext_vector_type(8))) float v8f;
typedef __attribute__((ext_vector_type(4))) int   v4i;

#define SI 256
#define VI 32
#define SE 64
#define VE 4
#define SO 256
#define VO 32
#define IN_S 576          // 2*SI + SE
#define IN_V 68           // 2*VI + VE
#define HDIM 68
#define KTOT 644          // IN_S + HDIM, divisible by 4
#define OUT_C 352         // SO + 3*VO
#define SA_STRIDE 660     // bank-conflict-free: (20*row)%64 distinct for row 0..15
#define SV_STRIDE 84      // same property, keeps b64 alignment
#define EPB 32            // edges per block = 2 WMMA M-tiles
#define NROWS 96          // 3 * EPB stacked vector rows
#define NTHREADS 256      // 8 waves of 32
#define EPSV 1e-8f

// Optional CDNA5 async global->LDS path (compile-safe fallback otherwise)
#if defined(__has_builtin)
# if __has_builtin(__builtin_amdgcn_global_load_async_to_lds_b128) && \
     __has_builtin(__builtin_amdgcn_s_wait_asynccnt)
#  define USE_ASYNC_LDS 1
# endif
#endif
#ifndef USE_ASYNC_LDS
# define USE_ASYNC_LDS 0
#endif

#if USE_ASYNC_LDS
// Builtin signature (from clang diagnostic): arg0 = v4i in addrspace(1) (global),
// arg1 = v4i in addrspace(3) (LDS), arg2 = imm offset, arg3 = cpol.
typedef __attribute__((address_space(1))) v4i v4i_glb;
typedef __attribute__((address_space(3))) v4i v4i_lds;
#endif

// ---------------------------------------------------------------------------
// Main edge-message kernel: gathers, 3 f32-WMMA GEMMs, atomic scatter-add.
// ---------------------------------------------------------------------------
__global__ __launch_bounds__(NTHREADS)
void gbp_edge_kernel(const float* __restrict__ node_s,
                     const float* __restrict__ node_v,
                     const float* __restrict__ edge_s,
                     const float* __restrict__ edge_v,
                     const float* __restrict__ wh,
                     const float* __restrict__ ws_w,
                     const float* __restrict__ ws_b,
                     const float* __restrict__ wv,
                     const long long* __restrict__ edge_index,
                     float* __restrict__ out,
                     float* __restrict__ cnt,
                     int E)
{
    __shared__ float sA[EPB * SA_STRIDE];     // [32][644] scalar cat (+vn)
    __shared__ float sV[NROWS * SV_STRIDE];   // v_cat stacked: row = dim*32 + edge
    __shared__ float sVH[NROWS * SV_STRIDE];  // vh stacked, same layout
    __shared__ int   s_src[EPB], s_dst[EPB];

    const int tid  = threadIdx.x;
    const int lane = tid & 31;
    const int wid  = tid >> 5;
    const int e0   = blockIdx.x * EPB;

    // ---- edge indices + per-edge count ----
    if (tid < EPB) {
        int eg = e0 + tid;
        int ec = eg < E ? eg : (E - 1);
        s_src[tid] = (int)edge_index[ec];
        int d      = (int)edge_index[(size_t)E + ec];
        s_dst[tid] = d;
        if (eg < E) atomicAdd(&cnt[d], 1.0f);
    }
    __syncthreads();

    // ---- gather scalar features [s_j | e_s | s_i] into sA ----
#if USE_ASYNC_LDS
    for (int i = tid; i < EPB * (IN_S / 4); i += NTHREADS) {
        int e = i / (IN_S / 4), c = (i - e * (IN_S / 4)) * 4;
        int eg = e0 + e; if (eg >= E) eg = E - 1;
        const float* g;
        if (c < SI)           g = node_s + (size_t)s_src[e] * SI + c;
        else if (c < SI + SE) g = edge_s + (size_t)eg * SE + (c - SI);
        else                  g = node_s + (size_t)s_dst[e] * SI + (c - SI - SE);
        __builtin_amdgcn_global_load_async_to_lds_b128(
            (v4i_glb*)g, (v4i_lds*)&sA[e * SA_STRIDE + c], 0, 0);
    }
#else
    for (int i = tid; i < EPB * IN_S; i += NTHREADS) {
        int e = i / IN_S, c = i - e * IN_S;
        int eg = e0 + e; if (eg >= E) eg = E - 1;
        float val;
        if (c < SI)           val = node_s[(size_t)s_src[e] * SI + c];
        else if (c < SI + SE) val = edge_s[(size_t)eg * SE + (c - SI)];
        else                  val = node_s[(size_t)s_dst[e] * SI + (c - SI - SE)];
        sA[e * SA_STRIDE + c] = val;
    }
#endif

    // ---- gather vector features, stacked [dim*32+e][channel], zero-padded ----
    for (int i = tid; i < NROWS * SV_STRIDE; i += NTHREADS) {
        int row = i / SV_STRIDE, c = i - row * SV_STRIDE;
        int d = row >> 5, e = row & 31;
        int eg = e0 + e; if (eg >= E) eg = E - 1;
        float val = 0.0f;
        if (c < VI)           val = node_v[(size_t)s_src[e] * (VI * 3) + c * 3 + d];
        else if (c < VI + VE) val = edge_v[(size_t)eg * (VE * 3) + (c - VI) * 3 + d];
        else if (c < IN_V)    val = node_v[(size_t)s_dst[e] * (VI * 3) + (c - VI - VE) * 3 + d];
        sV[i] = val;
    }

#if USE_ASYNC_LDS
    __builtin_amdgcn_s_wait_asynccnt(0);
#endif
    __syncthreads();

    const int lrow  = lane & 15;
    const int khalf = (lane >> 4) << 1;   // K sub-offset: 0 or 2
    const int mbase = (lane >> 4) << 3;   // C/D M base:   0 or 8

    // ---- GEMM1: vh[96][68] = v_cat[96][68] @ wh[68][68]  (30 tiles / 8 waves) ----
    for (int t = wid; t < 30; t += 8) {
        int mt = t / 5, nt = t - mt * 5;
        int cB  = nt * 16 + lrow;
        int cBc = cB < HDIM ? cB : 0;     // clamp in-bounds; garbage cols never read
        v8f acc = {};
        for (int k0 = 0; k0 < IN_V; k0 += 4) {
            int kk = k0 + khalf;
            v2f a, b;
            a.x = sV[(mt * 16 + lrow) * SV_STRIDE + kk];
            a.y = sV[(mt * 16 + lrow) * SV_STRIDE + kk + 1];
            b.x = wh[(size_t)kk * HDIM + cBc];
            b.y = wh[(size_t)(kk + 1) * HDIM + cBc];
            acc = __builtin_amdgcn_wmma_f32_16x16x4_f32(false, a, false, b,
                                                        (short)0, acc, false, false);
        }
        #pragma unroll
        for (int i = 0; i < 8; ++i)
            sVH[(mt * 16 + mbase + i) * SV_STRIDE + cB] = acc[i];
    }
    __syncthreads();

    // ---- vn = sqrt(max(sum_d vh^2, eps)) -> A-panel cols [576,644) ----
    for (int i = tid; i < EPB * HDIM; i += NTHREADS) {
        int e = i / HDIM, c = i - e * HDIM;
        float x0 = sVH[(          e) * SV_STRIDE + c];
        float x1 = sVH[(EPB     + e) * SV_STRIDE + c];
        float x2 = sVH[(2 * EPB + e) * SV_STRIDE + c];
        float s  = x0 * x0 + x1 * x1 + x2 * x2;
        sA[e * SA_STRIDE + IN_S + c] = sqrtf(fmaxf(s, EPSV));
    }
    __syncthreads();

    // ---- GEMM2 (dominant): s_msg[32][256] = sA[32][644] @ ws_w + ws_b ----
    // Each wave: 2 M-tiles x 2 N-tiles -> 4 independent accumulators;
    // every B fragment is read once per 32 edges, every A fragment feeds 2 WMMAs.
    {
        int cn0 = wid * 32 + lrow;
        int cn1 = cn0 + 16;
        float b0i = ws_b[cn0], b1i = ws_b[cn1];
        v8f acc00 = {b0i, b0i, b0i, b0i, b0i, b0i, b0i, b0i};
        v8f acc01 = {b1i, b1i, b1i, b1i, b1i, b1i, b1i, b1i};
        v8f acc10 = acc00, acc11 = acc01;
        #pragma unroll 7
        for (int k0 = 0; k0 < KTOT; k0 += 4) {
            int kk = k0 + khalf;
            v2f a0, a1, b0, b1;
            a0.x = sA[lrow * SA_STRIDE + kk];
            a0.y = sA[lrow * SA_STRIDE + kk + 1];
            a1.x = sA[(16 + lrow) * SA_STRIDE + kk];
            a1.y = sA[(16 + lrow) * SA_STRIDE + kk + 1];
            b0.x = ws_w[(size_t)kk * SO + cn0];
            b0.y = ws_w[(size_t)(kk + 1) * SO + cn0];
            b1.x = ws_w[(size_t)kk * SO + cn1];
            b1.y = ws_w[(size_t)(kk + 1) * SO + cn1];
            acc00 = __builtin_amdgcn_wmma_f32_16x16x4_f32(false, a0, false, b0,
                                                          (short)0, acc00, false, false);
            acc01 = __builtin_amdgcn_wmma_f32_16x16x4_f32(false, a0, false, b1,
                                                          (short)0, acc01, false, false);
            acc10 = __builtin_amdgcn_wmma_f32_16x16x4_f32(false, a1, false, b0,
                                                          (short)0, acc10, false, false);
            acc11 = __builtin_amdgcn_wmma_f32_16x16x4_f32(false, a1, false, b1,
                                                          (short)0, acc11, false, false);
        }
        #pragma unroll
        for (int i = 0; i < 8; ++i) {
            int ea = mbase + i, eb = 16 + mbase + i;
            if (e0 + ea < E) {
                atomicAdd(&out[(size_t)s_dst[ea] * OUT_C + cn0], acc00[i]);
                atomicAdd(&out[(size_t)s_dst[ea] * OUT_C + cn1], acc01[i]);
            }
            if (e0 + eb < E) {
                atomicAdd(&out[(size_t)s_dst[eb] * OUT_C + cn0], acc10[i]);
                atomicAdd(&out[(size_t)s_dst[eb] * OUT_C + cn1], acc11[i]);
            }
        }
    }

    // ---- GEMM3: v_msg[96][32] = vh[96][68] @ wv[68][32]  (12 tiles / 8 waves) ----
    for (int t = wid; t < 12; t += 8) {
        int mt = t >> 1, nt = t & 1;
        int cn = nt * 16 + lrow;
        v8f acc = {};
        for (int k0 = 0; k0 < HDIM; k0 += 4) {
            int kk = k0 + khalf;
            v2f a, b;
            a.x = sVH[(mt * 16 + lrow) * SV_STRIDE + kk];
            a.y = sVH[(mt * 16 + lrow) * SV_STRIDE + kk + 1];
            b.x = wv[(size_t)kk * VO + cn];
            b.y = wv[(size_t)(kk + 1) * VO + cn];
            acc = __builtin_amdgcn_wmma_f32_16x16x4_f32(false, a, false, b,
                                                        (short)0, acc, false, false);
        }
        int d = mt >> 1;                       // spatial dim
        #pragma unroll
        for (int i = 0; i < 8; ++i) {
            int e = (mt & 1) * 16 + mbase + i; // edge within block
            if (e0 + e < E)
                atomicAdd(&out[(size_t)s_dst[e] * OUT_C + SO + cn * 3 + d], acc[i]);
        }
    }
}

// ---------------------------------------------------------------------------
__global__ void gbp_zero_kernel(float* __restrict__ out, float* __restrict__ cnt,
                                int out_n, int n_nodes)
{
    int i = blockIdx.x * blockDim.x + threadIdx.x;
    if (i < out_n)   out[i] = 0.0f;
    if (i < n_nodes) cnt[i] = 0.0f;
}

__global__ void gbp_norm_kernel(float* __restrict__ out, const float* __restrict__ cnt,
                                int out_n)
{
    int i = blockIdx.x * blockDim.x + threadIdx.x;
    if (i < out_n) {
        int n = i / OUT_C;
        out[i] = out[i] / fmaxf(cnt[n], 1.0f);
    }
}

// ---------------------------------------------------------------------------
extern "C" void kernel_launch(void* const* d_in, const int* in_sizes, int n_in,
                              void* d_out, int out_size, void* d_ws, size_t ws_size,
                              hipStream_t stream)
{
    const float*     node_s = (const float*)d_in[0];
    const float*     node_v = (const float*)d_in[1];
    const float*     edge_s = (const float*)d_in[2];
    const float*     edge_v = (const float*)d_in[3];
    const float*     wh     = (const float*)d_in[4];
    const float*     ws_w   = (const float*)d_in[5];
    const float*     ws_b   = (const float*)d_in[6];
    const float*     wv     = (const float*)d_in[7];
    const long long* eidx   = (const long long*)d_in[8];

    float* out = (float*)d_out;
    float* cnt = (float*)d_ws;          // n_nodes floats of scratch

    const int n_nodes = in_sizes[0] / SI;
    const int n_edges = in_sizes[2] / SE;

    int zt = out_size > n_nodes ? out_size : n_nodes;
    gbp_zero_kernel<<<(zt + 255) / 256, 256, 0, stream>>>(out, cnt, out_size, n_nodes);

    int blocks = (n_edges + EPB - 1) / EPB;
    gbp_edge_kernel<<<blocks, NTHREADS, 0, stream>>>(node_s, node_v, edge_s, edge_v,
                                                     wh, ws_w, ws_b, wv, eidx,
                                                     out, cnt, n_edges);

    gbp_norm_kernel<<<(out_size + 255) / 256, 256, 0, stream>>>(out, cnt, out_size);
}